// ForexPrep_28905129902259
// MI455X (gfx1250) — compile-verified
//
#include <hip/hip_runtime.h>
#include <stdint.h>

#define GAP        7
#define TLEN       8192                 // T
#define PLEN       (TLEN - 1)           // 8191 p-values per row
#define OLEN       (TLEN - GAP)         // 8185 outputs per row
#define BLOCK      256
#define NWAVE      (BLOCK / 32)
#define PER_THREAD (TLEN / BLOCK)       // 32 strided elements per thread
#define CHUNKS     ((TLEN * 4) / (BLOCK * 16))   // 8 x b128 async loads per thread

typedef float v2f __attribute__((ext_vector_type(2)));   // native vector for NT store

__global__ __launch_bounds__(BLOCK)
void forex_prep_kernel(const float* __restrict__ x,
                       float* __restrict__ out)
{
    __shared__ float sx[TLEN];      // row of x, later overwritten with p
    __shared__ float red[NWAVE];    // cross-wave max scratch

    const int row = blockIdx.x;
    const int tid = threadIdx.x;
    const float* rowp = x + (size_t)row * TLEN;

    // ---- stage row into LDS via CDNA5 async global->LDS DMA path ----
    // LDS address comes from the vdst VGPR; flat LDS pointer's low 32 bits are the
    // LDS byte offset. NT hint: row is read once.
    uint32_t lds_base = (uint32_t)(uintptr_t)(&sx[0]);
#pragma unroll
    for (int k = 0; k < CHUNKS; ++k) {
        uint32_t boff = (uint32_t)((k * BLOCK + tid) * 16);
        uint32_t ldsa = lds_base + boff;
        asm volatile("global_load_async_to_lds_b128 %0, %1, %2 th:TH_LOAD_NT"
                     :: "v"(ldsa), "v"(boff), "s"(rowp)
                     : "memory");
    }
    asm volatile("s_wait_asynccnt 0" ::: "memory");
    __syncthreads();

    // ---- pass 1: p[j] = (x[j+1]-x[j]) * rcp(x[j]), reduce max|p| (== scale) ----
    float preg[PER_THREAD];
    float xreg[PER_THREAD];
    float amax = 0.0f;
#pragma unroll
    for (int k = 0; k < PER_THREAD; ++k) {
        int j = k * BLOCK + tid;          // j in [0, 8192)
        float a = sx[j];
        xreg[k] = a;
        float pj = 0.0f;
        if (j < PLEN) {
            float b = sx[j + 1];
            pj = (b - a) * __builtin_amdgcn_rcpf(a);   // v_rcp_f32, ~1 ulp
            amax = fmaxf(amax, fabsf(pj));
        }
        preg[k] = pj;
    }
    // wave32 butterfly reduction
#pragma unroll
    for (int m = 16; m > 0; m >>= 1)
        amax = fmaxf(amax, __shfl_xor(amax, m, 32));
    if ((tid & 31) == 0) red[tid >> 5] = amax;
    __syncthreads();    // red visible; all reads of sx (as x) complete

    // overwrite LDS row with p in place (no extra buffer needed)
#pragma unroll
    for (int k = 0; k < PER_THREAD; ++k) {
        int j = k * BLOCK + tid;
        sx[j] = preg[k];
    }
    float scale = red[0];
#pragma unroll
    for (int w = 1; w < NWAVE; ++w) scale = fmaxf(scale, red[w]);
    float inv = __builtin_amdgcn_rcpf((float)GAP * scale);
    __syncthreads();    // p ready in LDS

    // ---- pass 2: 7-tap window sum of p, interleave with x ----
    float* orow = out + (size_t)row * (size_t)OLEN * 2;
#pragma unroll
    for (int k = 0; k < PER_THREAD; ++k) {
        int i = k * BLOCK + tid;
        if (i < OLEN) {
            float s = sx[i];
#pragma unroll
            for (int g = 1; g < GAP; ++g) s += sx[i + g];
            v2f o;
            o.x = s * inv;       // ma
            o.y = xreg[k];       // x[:, :-GAP]
            // streaming output: bypass cache residency (TH_STORE_NT)
            __builtin_nontemporal_store(o, (v2f*)(orow + (size_t)i * 2));
        }
    }
}

extern "C" void kernel_launch(void* const* d_in, const int* in_sizes, int n_in,
                              void* d_out, int out_size, void* d_ws, size_t ws_size,
                              hipStream_t stream)
{
    const float* x = (const float*)d_in[0];
    float* out = (float*)d_out;
    const int nrows = in_sizes[0] / TLEN;   // 4096
    forex_prep_kernel<<<dim3(nrows), dim3(BLOCK), 0, stream>>>(x, out);
}